// HashGrid_22265110462894
// MI455X (gfx1250) — compile-verified
//
#include <hip/hip_runtime.h>
#include <cstdint>
#include <cstddef>

// ---------------------------------------------------------------------------
// HashGrid encode (instant-NGP / tcnn semantics), MI455X / gfx1250.
//
// N = 2,097,152 points, 16 levels x 2 feats, table 2^19 entries/level.
// Memory-bound: 256 MB output stores + ~1.3 GB of 8B gathers against a
// 64 MB table that fits in the 192 MB L2.  Strategy:
//   * one thread per point, full 16-level unroll -> 8 gathers in flight/level
//   * levels 0..5 dense stride indexing, 6..15 prime-hash (& (2^19-1))
//   * level-0 table cached in LDS via GLOBAL_LOAD_ASYNC_TO_LDS_B64
//     (ASYNCcnt); ds_load_b64 gathers afterwards
//   * NON-TEMPORAL b128 output stores + NT x loads so the 256 MB write
//     stream does not evict the L2-resident table
//   * persistent grid-stride so the 35 KB LDS preload amortizes
//   * global_prefetch_b8 of next tile's coords
// ---------------------------------------------------------------------------

#define AS1 __attribute__((address_space(1)))
#define AS3 __attribute__((address_space(3)))

typedef float v4f __attribute__((ext_vector_type(4)));
typedef int   v2i __attribute__((vector_size(2 * sizeof(int))));  // matches builtin param

namespace {

constexpr int      kLevels    = 16;
constexpr uint32_t kTable     = 1u << 19;
constexpr uint32_t kMask      = kTable - 1u;
constexpr uint32_t kP1        = 2654435761u;
constexpr uint32_t kP2        = 805459861u;
constexpr int      kDenseLvls = 6;       // levels 0..5: res^3 <= 2^19
constexpr int      kBlock     = 256;
constexpr int      kGrid      = 2048;    // persistent-ish grid-stride
// level-0 dense grid: indices reach 16 + 16*16 + 16*256 = 4368
constexpr int      kL0Entries = 4370;

constexpr double kPLS = 1.3542555593;    // 128^(1/16)

struct ScaleTab { float v[kLevels]; };
constexpr ScaleTab make_scales() {
  ScaleTab s{};
  double p = 1.0;
  for (int i = 0; i < kLevels; ++i) { s.v[i] = (float)(16.0 * p - 1.0); p *= kPLS; }
  return s;
}
constexpr ScaleTab kScale = make_scales();
constexpr uint32_t kRes[kLevels] = {16u, 22u, 30u, 40u, 54u, 73u,
                                    0u, 0u, 0u, 0u, 0u, 0u, 0u, 0u, 0u, 0u};

__device__ __forceinline__ void wait_asynccnt0() {
#if __has_builtin(__builtin_amdgcn_s_wait_asynccnt)
  __builtin_amdgcn_s_wait_asynccnt(0);
#else
  asm volatile("s_wait_asynccnt 0x0" ::: "memory");
#endif
}

__global__ __launch_bounds__(kBlock) void hashgrid_fwd(
    const float* __restrict__ x, const float* __restrict__ table,
    float* __restrict__ out, int n)
{
  __shared__ float2 lvl0[kL0Entries];    // ~35 KB LDS cache of level-0 table

  const float2* __restrict__ tbl = reinterpret_cast<const float2*>(table);

  // ---- cooperative preload of level-0 dense table into LDS ----------------
#if __has_builtin(__builtin_amdgcn_global_load_async_to_lds_b64)
  // gfx1250 async DMA path: no VGPR staging, tracked with ASYNCcnt.
  for (int i = threadIdx.x; i < kL0Entries; i += kBlock) {
    __builtin_amdgcn_global_load_async_to_lds_b64(
        (AS1 v2i*)(tbl + i), (AS3 v2i*)(&lvl0[i]), /*offset=*/0, /*cpol=*/0);
  }
  wait_asynccnt0();   // each wave retires its own async ops before signaling
  __syncthreads();
#else
  for (int i = threadIdx.x; i < kL0Entries; i += kBlock) lvl0[i] = tbl[i];
  __syncthreads();
#endif

  const int stride = gridDim.x * kBlock;
  for (int p = blockIdx.x * kBlock + threadIdx.x; p < n; p += stride) {
    const float* __restrict__ xp = x + (size_t)p * 3u;
    if (p + stride < n) {
      __builtin_prefetch(xp + (size_t)stride * 3u, 0, 0);  // global_prefetch_b8
    }
    // read-once coords: non-temporal so they don't occupy cache
    const float cx = __builtin_nontemporal_load(xp + 0);
    const float cy = __builtin_nontemporal_load(xp + 1);
    const float cz = __builtin_nontemporal_load(xp + 2);

    float acc[2 * kLevels];

#pragma unroll
    for (int l = 0; l < kLevels; ++l) {
      const float s = kScale.v[l];
      const float px = cx * s + 0.5f;
      const float py = cy * s + 0.5f;
      const float pz = cz * s + 0.5f;
      const float fx = floorf(px), fy = floorf(py), fz = floorf(pz);
      const float tx = px - fx, ty = py - fy, tz = pz - fz;
      const uint32_t ix = (uint32_t)fx;
      const uint32_t iy = (uint32_t)fy;
      const uint32_t iz = (uint32_t)fz;

      uint32_t c[8];
      if (l < kDenseLvls) {
        const uint32_t r = kRes[l], r2 = r * r;
        const uint32_t bx0 = ix,      bx1 = ix + 1u;
        const uint32_t by0 = iy * r,  by1 = by0 + r;
        const uint32_t bz0 = iz * r2, bz1 = bz0 + r2;
        c[0] = bx0 + by0 + bz0;  c[1] = bx1 + by0 + bz0;
        c[2] = bx0 + by1 + bz0;  c[3] = bx1 + by1 + bz0;
        c[4] = bx0 + by0 + bz1;  c[5] = bx1 + by0 + bz1;
        c[6] = bx0 + by1 + bz1;  c[7] = bx1 + by1 + bz1;
      } else {
        const uint32_t bx0 = ix,        bx1 = ix + 1u;
        const uint32_t by0 = iy * kP1,  by1 = by0 + kP1;  // strength-reduced
        const uint32_t bz0 = iz * kP2,  bz1 = bz0 + kP2;
        c[0] = (bx0 ^ by0 ^ bz0) & kMask;  c[1] = (bx1 ^ by0 ^ bz0) & kMask;
        c[2] = (bx0 ^ by1 ^ bz0) & kMask;  c[3] = (bx1 ^ by1 ^ bz0) & kMask;
        c[4] = (bx0 ^ by0 ^ bz1) & kMask;  c[5] = (bx1 ^ by0 ^ bz1) & kMask;
        c[6] = (bx0 ^ by1 ^ bz1) & kMask;  c[7] = (bx1 ^ by1 ^ bz1) & kMask;
      }

      // issue all 8 gathers before consuming -> max memory-level parallelism
      float2 f[8];
      if (l == 0) {
#pragma unroll
        for (int k = 0; k < 8; ++k) f[k] = lvl0[c[k]];          // ds_load_b64
      } else {
        const float2* __restrict__ t = tbl + (size_t)l * kTable;
#pragma unroll
        for (int k = 0; k < 8; ++k) f[k] = t[c[k]];             // global_load_b64
      }

      const float wx1 = tx, wx0 = 1.0f - tx;
      const float wy1 = ty, wy0 = 1.0f - ty;
      const float wz1 = tz, wz0 = 1.0f - tz;
      const float w00 = wy0 * wz0, w10 = wy1 * wz0;
      const float w01 = wy0 * wz1, w11 = wy1 * wz1;
      const float w[8] = { wx0 * w00, wx1 * w00, wx0 * w10, wx1 * w10,
                           wx0 * w01, wx1 * w01, wx0 * w11, wx1 * w11 };

      float a0 = 0.0f, a1 = 0.0f;
#pragma unroll
      for (int k = 0; k < 8; ++k) {
        a0 = fmaf(w[k], f[k].x, a0);
        a1 = fmaf(w[k], f[k].y, a1);
      }
      acc[2 * l]     = a0;
      acc[2 * l + 1] = a1;
    }

    // one point = exactly one 128B line; 8x NT b128 stores fill it completely
    // (non-temporal keeps the 256 MB write stream from evicting the table
    //  out of the 192 MB L2)
    v4f* __restrict__ op = reinterpret_cast<v4f*>(out + (size_t)p * 32u);
#pragma unroll
    for (int k = 0; k < 8; ++k) {
      v4f val = { acc[4 * k + 0], acc[4 * k + 1],
                  acc[4 * k + 2], acc[4 * k + 3] };
      __builtin_nontemporal_store(val, op + k);
    }
  }
}

}  // namespace

extern "C" void kernel_launch(void* const* d_in, const int* in_sizes, int n_in,
                              void* d_out, int out_size, void* d_ws, size_t ws_size,
                              hipStream_t stream) {
  (void)n_in; (void)out_size; (void)d_ws; (void)ws_size;
  const float* x     = (const float*)d_in[0];  // [N,3]
  const float* table = (const float*)d_in[1];  // [16, 2^19, 2]
  float* out         = (float*)d_out;          // [N,32]
  const int n = in_sizes[0] / 3;

  int blocks = kGrid;
  const int needed = (n + kBlock - 1) / kBlock;
  if (blocks > needed) blocks = needed;

  hashgrid_fwd<<<blocks, kBlock, 0, stream>>>(x, table, out, n);
}